// YoloLoss_42116449305098
// MI455X (gfx1250) — compile-verified
//
#include <hip/hip_runtime.h>
#include <cstdint>

// Problem constants (from reference)
#define HH 64
#define WW 64
#define GG (HH * WW)          // 4096 cells
#define BB 16
#define AA 5
#define CC 85
#define MM 50
#define NCLS 80
#define INV_RED 0.125f        // 1/8
#define THRESHF 0.6f
#define OBJ_SCALE_F 5.0f
#define NOOBJ_F 1.0f
#define GTILE 256
#define NBLK (BB * AA * (GG / GTILE))   // 1280 blocks

typedef float v2f __attribute__((ext_vector_type(2)));
typedef float v8f __attribute__((ext_vector_type(8)));

#if defined(__gfx1250__) && __has_builtin(__builtin_amdgcn_global_load_async_to_lds_b32)
#define USE_ASYNC_LDS 1
typedef __attribute__((address_space(1))) int g_int;
typedef __attribute__((address_space(3))) int l_int;
#endif

// ---------------------------------------------------------------------------
// Main fused kernel: one block = 256 cells of one (b, a) slice.
// Produces 3 partial sums per block: [coord, conf, cls] -> partial[c*NBLK+blk]
// ---------------------------------------------------------------------------
__global__ __launch_bounds__(GTILE) void yolo_main(
    const float* __restrict__ cur,   // (B, A, C, G)
    const float* __restrict__ gt,    // (B, M, 5)
    const float* __restrict__ gtv,   // (B, M)
    const float* __restrict__ anc,   // (A, 2)
    float* __restrict__ partial)     // (3, NBLK)
{
    __shared__ float  s_chan[5][GTILE];
    __shared__ float4 s_box[MM];          // {x1, y1, x2, y2}  one ds_load_b128
    __shared__ float2 s_ai[MM];           // {area, int_as_float(idx)} one ds_load_b64
    __shared__ float  s_tx[MM], s_ty[MM], s_tw[MM], s_th[MM], s_tcls[MM];
    __shared__ float  s_red[3][GTILE];

    const int tid = threadIdx.x;
    const int b   = blockIdx.z;
    const int a   = blockIdx.y;
    const int g0  = blockIdx.x * GTILE;
    const int g   = g0 + tid;

    const float* gbase = cur + ((size_t)(b * AA + a) * CC) * GG + g0;

    // ---- Stage the 5 needed channel rows (x,y,w,h,conf) into LDS ----------
#ifdef USE_ASYNC_LDS
    #pragma unroll
    for (int c = 0; c < 5; ++c) {
        __builtin_amdgcn_global_load_async_to_lds_b32(
            (g_int*)(gbase + (size_t)c * GG + tid),
            (l_int*)&s_chan[c][tid],
            0, 0);
    }
#if __has_builtin(__builtin_amdgcn_s_wait_asynccnt)
    __builtin_amdgcn_s_wait_asynccnt(0);
#else
    asm volatile("s_wait_asynccnt 0x0" ::: "memory");
#endif
#else
    #pragma unroll
    for (int c = 0; c < 5; ++c)
        s_chan[c][tid] = gbase[(size_t)c * GG + tid];
#endif

    // ---- Per-gt derived data (threads 0..49), redundant per block ----------
    if (tid < MM) {
        const int m = tid;
        const float* g5 = gt + ((size_t)b * MM + m) * 5;
        const float gx = g5[0], gy = g5[1], gwid = g5[2], ghei = g5[3], gcls = g5[4];
        const bool valid = gtv[(size_t)b * MM + m] > 0.0f;
        const float gcx = (gx + 0.5f * gwid) * INV_RED;
        const float gcy = (gy + 0.5f * ghei) * INV_RED;
        const float gw  = gwid * INV_RED;
        const float gh  = ghei * INV_RED;
        // best anchor by anchor-space IoU (argmax, first-max wins)
        int best = 0; float bestv = -1.0f;
        #pragma unroll
        for (int aa = 0; aa < AA; ++aa) {
            const float aw = anc[2 * aa], ah = anc[2 * aa + 1];
            const float inter = fminf(gw, aw) * fminf(gh, ah);
            const float uni   = gw * gh + aw * ah - inter;
            const float r     = inter / uni;
            if (r > bestv) { bestv = r; best = aa; }
        }
        int gi = (int)gcx; gi = gi < 0 ? 0 : (gi > WW - 1 ? WW - 1 : gi);
        int gj = (int)gcy; gj = gj < 0 ? 0 : (gj > HH - 1 ? HH - 1 : gj);
        const int cell = gj * WW + gi;
        const int idx  = valid ? (best * GG + cell) : -1;
        if (valid) {
            s_box[m] = make_float4(gcx - 0.5f * gw, gcy - 0.5f * gh,
                                   gcx + 0.5f * gw, gcy + 0.5f * gh);
            s_ai[m]  = make_float2(gw * gh, __int_as_float(idx));
        } else {
            // degenerate box => intersection always 0 => iou 0 (matches where(valid))
            s_box[m] = make_float4(1e30f, 1e30f, -1e30f, -1e30f);
            s_ai[m]  = make_float2(0.0f, __int_as_float(-1));
        }
        const float awb = anc[2 * best], ahb = anc[2 * best + 1];
        s_tx[m] = gcx - (float)gi;
        s_ty[m] = gcy - (float)gj;
        s_tw[m] = logf(fmaxf(gw, 1.0f) / awb);
        s_th[m] = logf(fmaxf(gh, 1.0f) / ahb);
        s_tcls[m] = gcls;
    }
    __syncthreads();

    // ---- Per-cell work -----------------------------------------------------
    const float rx = s_chan[0][tid], ry = s_chan[1][tid];
    const float rw = s_chan[2][tid], rh = s_chan[3][tid];
    const float rc = s_chan[4][tid];

    const float sx = 1.0f / (1.0f + expf(-rx));     // sigmoid x  (coord[0])
    const float sy = 1.0f / (1.0f + expf(-ry));     // sigmoid y  (coord[1])
    const float ew = expf(rw);                      // exp w      (coord[2])
    const float eh = expf(rh);                      // exp h      (coord[3])

    const float linx = (float)(g & (WW - 1));
    const float liny = (float)(g >> 6);
    const float px = sx + linx;
    const float py = sy + liny;
    // NOTE: reference applies exp() to the already-exp'ed wh (double exp). Replicate.
    const float pw = expf(ew) * anc[2 * a];
    const float ph = expf(eh) * anc[2 * a + 1];
    const float px1 = px - 0.5f * pw, px2 = px + 0.5f * pw;
    const float py1 = py - 0.5f * ph, py2 = py + 0.5f * ph;
    const float parea = pw * ph;                    // > 0 always => union > 0

    const int myflat = a * GG + g;
    bool  ignore = false;
    int   objm = -1;
    float obj_i = 0.0f, obj_u = 1.0f;

    #pragma unroll 2
    for (int m = 0; m < MM; ++m) {
        const float4 bx = s_box[m];                  // ds_load_b128 (broadcast)
        const float2 ai = s_ai[m];                   // ds_load_b64  (broadcast)
        float dx = fminf(px2, bx.z) - fmaxf(px1, bx.x);
        float dy = fminf(py2, bx.w) - fmaxf(py1, bx.y);
        dx = fmaxf(dx, 0.0f);
        dy = fmaxf(dy, 0.0f);
        const float inter = dx * dy;
        const float uni   = parea + ai.x - inter;
        ignore = ignore || (inter > THRESHF * uni);  // iou > 0.6, division-free
        if (__float_as_int(ai.y) == myflat) { objm = m; obj_i = inter; obj_u = uni; }
    }

    float cmask = ignore ? 0.0f : NOOBJ_F;
    float tconf = 0.0f;
    float l_coord = 0.0f, l_cls = 0.0f;

    if (objm >= 0) {                                // object cell (~800 of 327680)
        cmask = OBJ_SCALE_F;
        tconf = obj_i / obj_u;                      // iou_best
        const float d0 = sx - s_tx[objm];
        const float d1 = sy - s_ty[objm];
        const float d2 = ew - s_tw[objm];
        const float d3 = eh - s_th[objm];
        l_coord = d0 * d0 + d1 * d1 + d2 * d2 + d3 * d3;

        // sparse class cross-entropy: log_softmax over 80 strided logits
        const float* clsbase = cur + (((size_t)(b * AA + a) * CC) + 5) * GG + g;
        #pragma unroll 8
        for (int k = 0; k < NCLS; ++k)
            __builtin_prefetch(clsbase + (size_t)k * GG, 0, 0); // global_prefetch_b8
        float mx = -1e30f;
        for (int k = 0; k < NCLS; ++k)
            mx = fmaxf(mx, clsbase[(size_t)k * GG]);
        float se = 0.0f;
        for (int k = 0; k < NCLS; ++k)
            se += expf(clsbase[(size_t)k * GG] - mx);
        int t = (int)s_tcls[objm];
        t = t < 0 ? 0 : (t > NCLS - 1 ? NCLS - 1 : t);
        const float ce = -(clsbase[(size_t)t * GG] - mx - logf(se));
        l_cls = 2.0f * ce;                          // CLASS_SCALE(=1) * 2
    }

    const float sc = 1.0f / (1.0f + expf(-rc));
    const float dc = sc - tconf;
    const float l_conf = dc * dc * cmask;           // (diff*sqrt(mask))^2

    // ---- Deterministic block tree-reduce, all 3 components in one tree -----
    const int blin = (blockIdx.z * AA + blockIdx.y) * (GG / GTILE) + blockIdx.x;
    __syncthreads();
    s_red[0][tid] = l_coord;
    s_red[1][tid] = l_conf;
    s_red[2][tid] = l_cls;
    __syncthreads();
    for (int off = GTILE / 2; off > 0; off >>= 1) {
        if (tid < off) {
            s_red[0][tid] += s_red[0][tid + off];
            s_red[1][tid] += s_red[1][tid + off];
            s_red[2][tid] += s_red[2][tid + off];
        }
        __syncthreads();
    }
    if (tid < 3) partial[(size_t)tid * NBLK + blin] = s_red[tid][0];
}

// ---------------------------------------------------------------------------
// Final reduce: single wave32. Lane-strided serial sums, then one
// V_WMMA_F32_16X16X4_F32 (B = ones) reduces the 32 lane partials per component.
// Deterministic; exact fp32 throughout.
// ---------------------------------------------------------------------------
__global__ __launch_bounds__(32) void yolo_reduce(
    const float* __restrict__ partial, float* __restrict__ out)
{
    const int lane = threadIdx.x;
    float res[3];
    #pragma unroll
    for (int c = 0; c < 3; ++c) {
        float s = 0.0f;
        for (int i = lane; i < NBLK; i += 32)
            s += partial[(size_t)c * NBLK + i];
#if __has_builtin(__builtin_amdgcn_wmma_f32_16x16x4_f32)
        // A(16x4): lane L holds {s, 0}; rows m get contributions from lane m
        // (K=0..1) and lane m+16 (K=2..3). With B = ones, D[m][n] = rowsum(A,m).
        v2f aV; aV[0] = s;    aV[1] = 0.0f;
        v2f bV; bV[0] = 1.0f; bV[1] = 1.0f;
        v8f cA = {};
        v8f d = __builtin_amdgcn_wmma_f32_16x16x4_f32(
            false, aV, false, bV, (short)0, cA, false, false);
        float t = d[0] + d[1] + d[2] + d[3] + d[4] + d[5] + d[6] + d[7];
        // lanes 0..15 hold sum of rows 0..7; lanes 16..31 hold rows 8..15
        res[c] = __shfl(t, 0, 32) + __shfl(t, 16, 32);
#else
        float t = s;
        for (int off = 16; off > 0; off >>= 1) t += __shfl_down(t, off, 32);
        res[c] = __shfl(t, 0, 32);
#endif
    }
    if (lane == 0) {
        const float inv_b = 1.0f / (float)BB;
        const float lc = res[0] * inv_b;   // loss_coord
        const float lf = res[1] * inv_b;   // loss_conf
        const float ls = res[2] * inv_b;   // loss_cls
        out[0] = lc + lf + ls;             // loss_tot
        out[1] = lc;
        out[2] = lf;
        out[3] = ls;
    }
}

extern "C" void kernel_launch(void* const* d_in, const int* in_sizes, int n_in,
                              void* d_out, int out_size, void* d_ws, size_t ws_size,
                              hipStream_t stream) {
    (void)in_sizes; (void)n_in; (void)out_size; (void)ws_size;
    const float* cur = (const float*)d_in[0];   // (16,5,85,4096)
    const float* gt  = (const float*)d_in[1];   // (16,50,5)
    const float* gtv = (const float*)d_in[2];   // (16,50)
    const float* anc = (const float*)d_in[3];   // (5,2)
    float* out       = (float*)d_out;           // 4 floats
    float* partial   = (float*)d_ws;            // 3*1280 floats

    dim3 grid(GG / GTILE, AA, BB);              // (16, 5, 16) = 1280 blocks
    yolo_main<<<grid, GTILE, 0, stream>>>(cur, gt, gtv, anc, partial);
    yolo_reduce<<<1, 32, 0, stream>>>(partial, out);
}